// SAGPool_89060441850425
// MI455X (gfx1250) — compile-verified
//
#include <hip/hip_runtime.h>
#include <hip/hip_bf16.h>

// ---------------- problem constants (match reference) ----------------
#define NB   64          // graphs
#define NN   2048        // nodes per graph
#define ND   128         // feature dim
#define NK   1024        // kept nodes per graph (RATIO*N)
#define NTOT (NB * NN)   // 131072 total nodes
#define EPG  (NN * 32)   // 65536 edges per graph
#define SPLIT 8          // edge-slices per graph

typedef __attribute__((ext_vector_type(2))) float v2f;
typedef __attribute__((ext_vector_type(8))) float v8f;

__device__ __forceinline__ unsigned umax1(unsigned d) { return d < 1u ? 1u : d; }

// ---------------- k0: zero accumulators ----------------
__global__ void zero_kernel(unsigned* p, int n) {
    int i = blockIdx.x * 256 + threadIdx.x;
    if (i < n) p[i] = 0u;
}

// ---------------- k1: degree counting (LDS bins, atomic flush) ----------------
__global__ void degree_kernel(const int* __restrict__ src, const int* __restrict__ dst,
                              unsigned* __restrict__ outdeg, unsigned* __restrict__ indeg) {
    __shared__ unsigned lo[NN];
    __shared__ unsigned li[NN];
    const int g  = blockIdx.x >> 3;    // graph
    const int sl = blockIdx.x & (SPLIT - 1);
    for (int i = threadIdx.x; i < NN; i += 256) { lo[i] = 0u; li[i] = 0u; }
    __syncthreads();
    const int base = g * EPG + sl * (EPG / SPLIT);
    for (int e = threadIdx.x; e < EPG / SPLIT; e += 256) {
        int sle = src[base + e] & (NN - 1);   // local id (N is a power of two)
        int dle = dst[base + e] & (NN - 1);
        atomicAdd(&lo[sle], 1u);
        atomicAdd(&li[dle], 1u);
    }
    __syncthreads();
    for (int i = threadIdx.x; i < NN; i += 256) {
        if (lo[i]) atomicAdd(&outdeg[g * NN + i], lo[i]);
        if (li[i]) atomicAdd(&indeg[g * NN + i], li[i]);
    }
}

// ---------------- k2: projection s = feature @ W via V_WMMA_F32_16X16X4_F32 ----------------
// One wave computes a 16-row tile. A = 16x4 f32 tile of feature, B = 4x16 with W[k]
// broadcast across columns -> every column of the 16x16 f32 accumulator equals the
// per-row dot product. Exact f32 precision (matches the f32 reference).
__global__ void project_wmma_kernel(const float* __restrict__ feat,
                                    const float* __restrict__ W,
                                    float* __restrict__ s, int nrows) {
    const int lane = threadIdx.x & 31;
    const int wave = threadIdx.x >> 5;
    const int tile = (blockIdx.x * 8 + wave) * 16;   // 16 rows per wave
    if (tile >= nrows) return;                        // wave-uniform: EXEC stays all-ones
    const int half = lane >> 4;       // 0: K=0,1 / rows 0-7 of D ; 1: K=2,3 / rows 8-15
    const int m    = lane & 15;       // row within tile (A fragment)
    const int kk   = half * 2;
    const float* rowp = feat + (size_t)(tile + m) * ND;
    v8f c = {};
    for (int k = 0; k < ND; k += 4) {
        v2f a, b;
        a.x = rowp[k + kk];
        a.y = rowp[k + kk + 1];
        b.x = W[k + kk];              // B row K broadcast across all 16 columns
        b.y = W[k + kk + 1];
        c = __builtin_amdgcn_wmma_f32_16x16x4_f32(
                /*neg_a=*/false, a, /*neg_b=*/false, b,
                /*c_mod=*/(short)0, c, /*reuse_a=*/false, /*reuse_b=*/false);
    }
    // Column 0 of D: lane 0 holds rows tile..tile+7 in c[0..7]; lane 16 rows tile+8..+15.
    if (m == 0) {
        float* o = s + tile + half * 8;
        float4 lo4 = make_float4(c[0], c[1], c[2], c[3]);
        float4 hi4 = make_float4(c[4], c[5], c[6], c[7]);
        ((float4*)o)[0] = lo4;
        ((float4*)o)[1] = hi4;
    }
}

// ---------------- k3: scatter score[dst] += s[src]*rsqrt(outdeg[src]) ----------------
__global__ void scatter_kernel(const int* __restrict__ src, const int* __restrict__ dst,
                               const float* __restrict__ s, const unsigned* __restrict__ outdeg,
                               float* __restrict__ score) {
    __shared__ float sv[NN];
    __shared__ float acc[NN];
    const int g  = blockIdx.x >> 3;
    const int sl = blockIdx.x & (SPLIT - 1);
    for (int i = threadIdx.x; i < NN; i += 256) {
        unsigned d = umax1(outdeg[g * NN + i]);
        sv[i]  = s[g * NN + i] * rsqrtf((float)d);
        acc[i] = 0.0f;
    }
    __syncthreads();
    const int base = g * EPG + sl * (EPG / SPLIT);
    for (int e = threadIdx.x; e < EPG / SPLIT; e += 256) {
        int sle = src[base + e] & (NN - 1);
        int dle = dst[base + e] & (NN - 1);
        atomicAdd(&acc[dle], sv[sle]);
    }
    __syncthreads();
    for (int i = threadIdx.x; i < NN; i += 256)
        if (acc[i] != 0.0f) atomicAdd(&score[g * NN + i], acc[i]);
}

// ---------------- k4: normalize + bias + tanh + per-block max ----------------
__global__ void normalize_kernel(const unsigned* __restrict__ indeg,
                                 const float* __restrict__ bptr,
                                 float* __restrict__ score, float* __restrict__ t,
                                 float* __restrict__ pmax, int n) {
    __shared__ float red[256];
    int i = blockIdx.x * 256 + threadIdx.x;
    float v = -__builtin_inff();
    if (i < n) {
        unsigned d = umax1(indeg[i]);
        float sc = score[i] * rsqrtf((float)d) + bptr[0];
        score[i] = sc;
        t[i] = tanhf(sc);
        v = sc;
    }
    red[threadIdx.x] = v;
    __syncthreads();
    for (int off = 128; off > 0; off >>= 1) {
        if (threadIdx.x < off) red[threadIdx.x] = fmaxf(red[threadIdx.x], red[threadIdx.x + off]);
        __syncthreads();
    }
    if (threadIdx.x == 0) pmax[blockIdx.x] = red[0];
}

// ---------------- k5/k7: single-block reductions ----------------
__global__ void reduce_max_kernel(const float* __restrict__ p, float* __restrict__ out, int np) {
    __shared__ float red[256];
    float v = -__builtin_inff();
    for (int i = threadIdx.x; i < np; i += 256) v = fmaxf(v, p[i]);
    red[threadIdx.x] = v;
    __syncthreads();
    for (int off = 128; off > 0; off >>= 1) {
        if (threadIdx.x < off) red[threadIdx.x] = fmaxf(red[threadIdx.x], red[threadIdx.x + off]);
        __syncthreads();
    }
    if (threadIdx.x == 0) out[0] = red[0];
}

__global__ void sumexp_kernel(const float* __restrict__ score, const float* __restrict__ fmax,
                              float* __restrict__ psum, int n) {
    __shared__ float red[256];
    int i = blockIdx.x * 256 + threadIdx.x;
    float v = (i < n) ? expf(score[i] - fmax[0]) : 0.0f;
    red[threadIdx.x] = v;
    __syncthreads();
    for (int off = 128; off > 0; off >>= 1) {
        if (threadIdx.x < off) red[threadIdx.x] += red[threadIdx.x + off];
        __syncthreads();
    }
    if (threadIdx.x == 0) psum[blockIdx.x] = red[0];
}

__global__ void reduce_sum_kernel(const float* __restrict__ p, float* __restrict__ out, int np) {
    __shared__ float red[256];
    float v = 0.0f;
    for (int i = threadIdx.x; i < np; i += 256) v += p[i];
    red[threadIdx.x] = v;
    __syncthreads();
    for (int off = 128; off > 0; off >>= 1) {
        if (threadIdx.x < off) red[threadIdx.x] += red[threadIdx.x + off];
        __syncthreads();
    }
    if (threadIdx.x == 0) out[0] = red[0];
}

// ---------------- k8: exact per-graph top-K via stable rank (one block per graph) ----------
// rank_i = #{ s_j > s_i } + #{ s_j == s_i && j < i }  -> matches jax.lax.top_k order exactly.
// Complement position of unselected i = #{ j < i : unselected } (ascending ids).
__global__ void topk_kernel(const float* __restrict__ score,
                            int* __restrict__ perm_i, int* __restrict__ permc_i,
                            float* __restrict__ out_perm, float* __restrict__ out_permc) {
    __shared__ float sc[NN];
    __shared__ int   flags[NN];
    const int g = blockIdx.x;
    for (int i = threadIdx.x; i < NN; i += 256) sc[i] = score[g * NN + i];
    __syncthreads();
    for (int i = threadIdx.x; i < NN; i += 256) {
        float v = sc[i];
        int cnt = 0;
        for (int j = 0; j < NN; ++j) {
            float u = sc[j];
            cnt += (u > v) || ((u == v) && (j < i));
        }
        int sel = (cnt < NK);
        flags[i] = sel;
        if (sel) {
            int gi = g * NN + i;
            perm_i[g * NK + cnt]   = gi;
            out_perm[g * NK + cnt] = (float)gi;
        }
    }
    __syncthreads();
    for (int i = threadIdx.x; i < NN; i += 256) {
        if (!flags[i]) {
            int c2 = 0;
            for (int j = 0; j < i; ++j) c2 += (flags[j] ^ 1);
            int gi = g * NN + i;
            permc_i[g * (NN - NK) + c2]   = gi;
            out_permc[g * (NN - NK) + c2] = (float)gi;
        }
    }
}

// ---------------- k9: scaled row gathers (one wave per 512B row, float4/lane) ------------
__global__ void gather_scale_kernel(const float* __restrict__ feat, const float* __restrict__ t,
                                    const int* __restrict__ perm, float* __restrict__ out,
                                    int rows) {
    const int lane = threadIdx.x & 31;
    const int wave = threadIdx.x >> 5;
    const int r = blockIdx.x * 8 + wave;
    if (r >= rows) return;
    const int srcRow = perm ? perm[r] : r;
    const float sc = t[srcRow];
    float4 v = ((const float4*)(feat + (size_t)srcRow * ND))[lane];
    float4 o = make_float4(v.x * sc, v.y * sc, v.z * sc, v.w * sc);
    ((float4*)(out + (size_t)r * ND))[lane] = o;
}

// ---------------- k10: softmax write ----------------
__global__ void softmax_kernel(const float* __restrict__ score, const float* __restrict__ fmax,
                               const float* __restrict__ fsum, float* __restrict__ out, int n) {
    int i = blockIdx.x * 256 + threadIdx.x;
    if (i < n) out[i] = expf(score[i] - fmax[0]) / fsum[0];
}

extern "C" void kernel_launch(void* const* d_in, const int* in_sizes, int n_in,
                              void* d_out, int out_size, void* d_ws, size_t ws_size,
                              hipStream_t stream) {
    const float* feature = (const float*)d_in[0];   // [NTOT, ND]
    const float* W       = (const float*)d_in[1];   // [ND, 1]
    const float* b       = (const float*)d_in[2];   // [1]
    const int*   src     = (const int*)d_in[3];     // [NB*EPG]
    const int*   dst     = (const int*)d_in[4];     // [NB*EPG]

    const int n = NTOT;

    // ---- workspace layout (floats/u32) ----
    float*    ws_f    = (float*)d_ws;
    float*    s_arr   = ws_f;                      // n
    unsigned* outdeg  = (unsigned*)(ws_f + n);     // n
    unsigned* indeg   = (unsigned*)(ws_f + 2 * n); // n
    float*    score   = ws_f + 3 * (size_t)n;      // n
    float*    t_arr   = ws_f + 4 * (size_t)n;      // n
    int*      perm_i  = (int*)(ws_f + 5 * (size_t)n);            // NB*NK
    int*      permc_i = perm_i + NB * NK;                        // NB*(NN-NK)
    float*    pmax    = ws_f + 6 * (size_t)n;      // 512
    float*    fmax    = pmax + 512;                // 1
    float*    psum    = fmax + 1;                  // 512
    float*    fsum    = psum + 512;                // 1

    // ---- output layout (all float, concatenated in reference return order) ----
    float* out_f     = (float*)d_out;
    float* out_dist  = out_f;                                        // NB*NK*ND
    float* out_com   = out_dist + (size_t)NB * NK * ND;              // NB*(NN-NK)*ND
    float* out_full  = out_com  + (size_t)NB * (NN - NK) * ND;       // NTOT*ND
    float* out_perm  = out_full + (size_t)NTOT * ND;                 // NB*NK
    float* out_permc = out_perm + (size_t)NB * NK;                   // NB*(NN-NK)
    float* out_sm    = out_permc + (size_t)NB * (NN - NK);           // NTOT

    // k0: zero outdeg/indeg/score (contiguous 3n words starting at outdeg)
    zero_kernel<<<(3 * n + 255) / 256, 256, 0, stream>>>(outdeg, 3 * n);

    // k1: degrees
    degree_kernel<<<NB * SPLIT, 256, 0, stream>>>(src, dst, outdeg, indeg);

    // k2: projection via WMMA (16 rows/wave, 8 waves/block -> 128 rows/block)
    project_wmma_kernel<<<n / 128, 256, 0, stream>>>(feature, W, s_arr, n);

    // k3: edge scatter with LDS accumulation
    scatter_kernel<<<NB * SPLIT, 256, 0, stream>>>(src, dst, s_arr, outdeg, score);

    // k4: normalize + bias + tanh + partial max
    const int nb512 = (n + 255) / 256;  // 512
    normalize_kernel<<<nb512, 256, 0, stream>>>(indeg, b, score, t_arr, pmax, n);

    // k5: final max
    reduce_max_kernel<<<1, 256, 0, stream>>>(pmax, fmax, nb512);

    // k6: partial sum of exp
    sumexp_kernel<<<nb512, 256, 0, stream>>>(score, fmax, psum, n);

    // k7: final sum
    reduce_sum_kernel<<<1, 256, 0, stream>>>(psum, fsum, nb512);

    // k8: per-graph exact top-K + complement, writes perm outputs too
    topk_kernel<<<NB, 256, 0, stream>>>(score, perm_i, permc_i, out_perm, out_permc);

    // k9: three scaled gathers (wave per row)
    gather_scale_kernel<<<(NB * NK + 7) / 8, 256, 0, stream>>>(feature, t_arr, perm_i, out_dist, NB * NK);
    gather_scale_kernel<<<(NB * (NN - NK) + 7) / 8, 256, 0, stream>>>(feature, t_arr, permc_i, out_com, NB * (NN - NK));
    gather_scale_kernel<<<(n + 7) / 8, 256, 0, stream>>>(feature, t_arr, nullptr, out_full, n);

    // k10: global softmax output
    softmax_kernel<<<nb512, 256, 0, stream>>>(score, fmax, fsum, out_sm, n);
}